// BundleAdjustment_40063454937165
// MI455X (gfx1250) — compile-verified
//
#include <hip/hip_runtime.h>

// Bundle-adjustment residuals for MI455X (gfx1250, wave32).
// Layout of d_out (float32, 12,640,256 elems):
//   [0,                 2*E_NUM)                       residual_proj  (err_r, err_t interleaved)
//   [2*E_NUM,           2*E_NUM + 7*P_NUM)             residual_pose
//   [2*E_NUM + 7*P_NUM, 2*E_NUM + 7*P_NUM + E_NUM)     residual_elev

#define BA_E_NUM 4194304
#define BA_P_NUM 8192

typedef float __attribute__((ext_vector_type(2))) f32x2;

// WGP-scope prefetch (scope field = 0): fills ALL cache levels including the
// WGP cache. __builtin_prefetch can only reach SE/SYS scope, which per the
// ISA prefetch table does NOT fill the WGP$ — useless for a GL2-resident
// table. Inline asm with no scope modifier encodes scope 0.
__device__ __forceinline__ void prefetch_wgp(const void* p) {
    asm volatile("global_prefetch_b8 %0, off" :: "v"(p) : "memory");
}
__device__ __forceinline__ void prefetch_wgp_off24(const void* p) {
    asm volatile("global_prefetch_b8 %0, off offset:24" :: "v"(p) : "memory");
}

__device__ __forceinline__ void quat_rotate(float qx, float qy, float qz, float qw,
                                            float& x, float& y, float& z) {
    // t = 2 * cross(u, v)
    float tx = 2.0f * (qy * z - qz * y);
    float ty = 2.0f * (qz * x - qx * z);
    float tz = 2.0f * (qx * y - qy * x);
    // v + w*t + cross(u, t)
    float rx = x + qw * tx + (qy * tz - qz * ty);
    float ry = y + qw * ty + (qz * tx - qx * tz);
    float rz = z + qw * tz + (qx * ty - qy * tx);
    x = rx; y = ry; z = rz;
}

__global__ __launch_bounds__(256) void ba_residual_kernel(
    const float* __restrict__ poses,            // (P_NUM, 7)
    const float* __restrict__ init_poses,       // (P_NUM, 7)
    const float* __restrict__ patch_coords,     // (E_NUM, 2)  r, theta
    const float* __restrict__ elevation,        // (E_NUM,)
    const float* __restrict__ init_elevation,   // (E_NUM,)
    const float* __restrict__ target_coords,    // (E_NUM, 2)
    const int*   __restrict__ src_idx,          // (E_NUM,)
    const int*   __restrict__ tgt_idx,          // (E_NUM,)
    const int*   __restrict__ patch_idx,        // (E_NUM,)
    float* __restrict__ out)
{
    const int e = blockIdx.x * 256 + threadIdx.x;
    if (e >= BA_E_NUM) return;

    constexpr float R_RANGE  = 30.0f - 0.5f;   // R_MAX - R_MIN
    constexpr float BINS     = 512.0f;
    constexpr float BEAMS    = 512.0f;
    constexpr float FOV_H    = 2.0943951f;
    constexpr int   POSE_OFF = 2 * BA_E_NUM;
    constexpr int   ELEV_OFF = 2 * BA_E_NUM + 7 * BA_P_NUM;

    // -------- indices: read-once streams -> non-temporal (protect L2 for the
    // gather tables: patch_coords 32MB + elevation 16MB + poses 224KB) --------
    const int si = __builtin_nontemporal_load(src_idx + e);
    const int ti = __builtin_nontemporal_load(tgt_idx + e);
    const int pi = __builtin_nontemporal_load(patch_idx + e);

    // Pose entries are 28-byte strided random gathers out of a 224KB
    // GL2-resident table. WGP-scope prefetch stages both entries into the
    // WGP$ while the transcendental block below executes.
    const float* __restrict__ Ps = poses + 7 * si;
    const float* __restrict__ Pt = poses + 7 * ti;
    prefetch_wgp(Ps);
    prefetch_wgp_off24(Ps);   // 28B entry may straddle a cacheline
    prefetch_wgp(Pt);
    prefetch_wgp_off24(Pt);

    // -------- gathered patch data (L2-resident tables, default hints) --------
    const f32x2 pc = *reinterpret_cast<const f32x2*>(patch_coords + 2 * (size_t)pi); // r, th
    const float ph = elevation[pi];

    // -------- polar -> cartesian (transcendental block; overlaps prefetch) ----
    float sph, cph, sth, cth;
    sincosf(ph,   &sph, &cph);
    sincosf(pc.y, &sth, &cth);
    const float rcp = pc.x * cph;
    float vx = rcp * cth;
    float vy = rcp * sth;
    float vz = pc.x * sph;

    // -------- source pose: rotate + translate --------
    const float stx = Ps[0], sty = Ps[1], stz = Ps[2];
    const float sqx = Ps[3], sqy = Ps[4], sqz = Ps[5], sqw = Ps[6];
    quat_rotate(sqx, sqy, sqz, sqw, vx, vy, vz);
    float gx = vx + stx;
    float gy = vy + sty;
    float gz = vz + stz;

    // -------- target pose: inverse transform --------
    const float ttx = Pt[0], tty = Pt[1], ttz = Pt[2];
    const float tqx = Pt[3], tqy = Pt[4], tqz = Pt[5], tqw = Pt[6];
    float dx = gx - ttx;
    float dy = gy - tty;
    float dz = gz - ttz;
    quat_rotate(-tqx, -tqy, -tqz, tqw, dx, dy, dz);

    // -------- cartesian -> polar (only rho, theta needed) --------
    const float rho   = sqrtf(dx * dx + dy * dy + dz * dz);
    const float theta = atan2f(dy, dx);

    const f32x2 tc = __builtin_nontemporal_load(
        reinterpret_cast<const f32x2*>(target_coords + 2 * (size_t)e));

    f32x2 err;
    err.x = (rho   - tc.x) / R_RANGE * BINS;
    err.y = (theta - tc.y) / FOV_H  * BEAMS;

    // Write-once outputs -> non-temporal stores (8B-aligned vector store for
    // the interleaved projection residual pair).
    __builtin_nontemporal_store(err, reinterpret_cast<f32x2*>(out + 2 * (size_t)e));

    // -------- streaming residuals folded into the same pass --------
    const float ie = __builtin_nontemporal_load(init_elevation + e);
    __builtin_nontemporal_store(elevation[e] - ie, out + ELEV_OFF + e);

    if (e < 7 * BA_P_NUM) {
        __builtin_nontemporal_store(poses[e] - init_poses[e], out + POSE_OFF + e);
    }
}

extern "C" void kernel_launch(void* const* d_in, const int* in_sizes, int n_in,
                              void* d_out, int out_size, void* d_ws, size_t ws_size,
                              hipStream_t stream) {
    (void)in_sizes; (void)n_in; (void)d_ws; (void)ws_size; (void)out_size;

    const float* poses          = (const float*)d_in[0];
    const float* init_poses     = (const float*)d_in[1];
    const float* patch_coords   = (const float*)d_in[2];
    const float* elevation      = (const float*)d_in[3];
    const float* init_elevation = (const float*)d_in[4];
    const float* target_coords  = (const float*)d_in[5];
    const int*   src_idx        = (const int*)d_in[6];
    const int*   tgt_idx        = (const int*)d_in[7];
    const int*   patch_idx      = (const int*)d_in[8];
    float* out                  = (float*)d_out;

    const int threads = 256;                       // 8 wave32 waves per block
    const int blocks  = BA_E_NUM / threads;        // 16384 blocks
    ba_residual_kernel<<<blocks, threads, 0, stream>>>(
        poses, init_poses, patch_coords, elevation, init_elevation,
        target_coords, src_idx, tgt_idx, patch_idx, out);
}